// SpatialExternalMemory_76699525972202
// MI455X (gfx1250) — compile-verified
//
#include <hip/hip_runtime.h>

// Pure-gather op: D[b, cell, :] = memory[clamp(gx+dx), clamp(gy+dy), :]
// for a (2w+1)^2 window. Zero FLOPs -> matrix units cannot help; this is a
// 23.3 TB/s HBM streaming problem (~392 MB total traffic, ~17 us floor).
//
// Strategy (MI455X / gfx1250, wave32):
//  - one 256-thread block (8 waves) per query
//  - 4 groups of 64 threads; each group copies one 1 KB window cell
//    (256 fp32 = 64 x float4) per iteration -> per-wave 512 B contiguous
//    bursts on both load and store (B128 ops)
//  - non-temporal B128 stores: the 196 MB write-once output should not
//    evict the read-side working set from the 192 MB L2
//  - global_prefetch_b8 on the next cell's address to hide gather latency

typedef float v4f __attribute__((ext_vector_type(4)));

__global__ __launch_bounds__(256) void SpatialExternalMemory_gather_kernel(
    const int* __restrict__ grid_input,   // [B, 2] (int32 per harness contract)
    const int* __restrict__ sw_ptr,       // [1] spatial_width
    const v4f* __restrict__ mem4,         // [NX*NY*H/4] float4 view of memory
    v4f* __restrict__ out4,               // [B*KK*H/4] float4 view of output
    int NX, int NY, int vecsPerCell, int KK)
{
    const int b = blockIdx.x;
    const int w = sw_ptr[0];
    const int K = 2 * w + 1;

    int gx = grid_input[2 * b + 0];
    int gy = grid_input[2 * b + 1];
    gx = min(max(gx, 0), NX - 1);
    gy = min(max(gy, 0), NY - 1);

    const int tid  = (int)threadIdx.x;
    const int grp  = tid >> 6;    // 4 groups of 64 threads (2 waves each)
    const int lane = tid & 63;

    const long long outBase = (long long)b * KK * vecsPerCell;

    for (int cell = grp; cell < KK; cell += 4) {
        // Decompose cell -> (cx, cy) within the K x K window.
        unsigned int cx = (unsigned int)cell / (unsigned int)K;
        unsigned int cy = (unsigned int)cell - cx * (unsigned int)K;
        int x = min(max(gx + (int)cx - w, 0), NX - 1);
        int y = min(max(gy + (int)cy - w, 0), NY - 1);
        const long long srcCell = ((long long)x * NY + y) * vecsPerCell;

        // Prefetch the next cell this group will handle (gfx1250
        // global_prefetch_b8; speculative, no counter cost).
        const int cellN = cell + 4;
        if (cellN < KK) {
            unsigned int cx2 = (unsigned int)cellN / (unsigned int)K;
            unsigned int cy2 = (unsigned int)cellN - cx2 * (unsigned int)K;
            int x2 = min(max(gx + (int)cx2 - w, 0), NX - 1);
            int y2 = min(max(gy + (int)cy2 - w, 0), NY - 1);
            __builtin_prefetch(
                (const void*)&mem4[((long long)x2 * NY + y2) * vecsPerCell + lane],
                /*rw=*/0, /*locality=*/1);
        }

        const long long dstCell = outBase + (long long)cell * vecsPerCell;
        // H = 256 -> vecsPerCell = 64 -> exactly one float4 per lane;
        // loop kept for generality in H.
        for (int vec = lane; vec < vecsPerCell; vec += 64) {
            v4f v = mem4[srcCell + vec];                     // global_load_b128
            __builtin_nontemporal_store(v, &out4[dstCell + vec]); // b128 NT store
        }
    }
}

extern "C" void kernel_launch(void* const* d_in, const int* in_sizes, int n_in,
                              void* d_out, int out_size, void* d_ws, size_t ws_size,
                              hipStream_t stream) {
    const int*   grid_input = (const int*)d_in[0];   // [B,2] int
    const int*   sw         = (const int*)d_in[1];   // [1] int (spatial_width)
    const float* memory     = (const float*)d_in[2]; // [NX,NY,H] fp32
    float*       out        = (float*)d_out;         // [B,KK,H] fp32

    const int B  = in_sizes[0] / 2;                  // 4096
    const int NX = 512;
    const int NY = 512;
    const int H  = in_sizes[2] / (NX * NY);          // 256
    const int vecsPerCell = H / 4;                   // 64 float4 per cell
    const int KK = out_size / (B * H);               // 49 window cells

    SpatialExternalMemory_gather_kernel<<<dim3(B), dim3(256), 0, stream>>>(
        grid_input, sw,
        (const v4f*)memory, (v4f*)out,
        NX, NY, vecsPerCell, KK);
}